// LayerStatistics_2138893713605
// MI455X (gfx1250) — compile-verified
//
#include <hip/hip_runtime.h>
#include <hip/hip_bf16.h>
#include <math.h>

typedef __attribute__((ext_vector_type(2))) float v2f;
typedef __attribute__((ext_vector_type(4))) float v4f;
typedef __attribute__((ext_vector_type(8))) float v8f;

// ---------------------------------------------------------------------------
// Wave32 sum reduction via V_WMMA_F32_16X16X4_F32.
// A-matrix layout (16x4 f32): VGPR0 lanes0-15 -> K=0, lanes16-31 -> K=2;
// VGPR1 lanes0-15 -> K=1, lanes16-31 -> K=3.  We place the lane partial in
// a.x and zero a.y, so A[m][0]=v_m, A[m][2]=v_{m+16}, K1/K3 = 0.
// With B = ones (4x16): D[m][n] = v_m + v_{m+16} for every n.
// D layout: VGPR r, lanes0-15 -> M=r, lanes16-31 -> M=r+8.  Hence
// t = sum(d[0..7]) covers M=0..7 (lanes<16) or M=8..15 (lanes>=16), and
// t + shfl_xor(t,16) = full 32-lane sum, broadcast to all lanes.
// Requires EXEC == all ones (call sites are divergence-free).
// ---------------------------------------------------------------------------
__device__ __forceinline__ float wave_sum32(float v) {
#if __has_builtin(__builtin_amdgcn_wmma_f32_16x16x4_f32)
    v2f a; a.x = v;    a.y = 0.0f;
    v2f b; b.x = 1.0f; b.y = 1.0f;
    v8f c = {};
    v8f d = __builtin_amdgcn_wmma_f32_16x16x4_f32(
        /*neg_a=*/false, a, /*neg_b=*/false, b,
        /*c_mod=*/(short)0, c, /*reuse_a=*/false, /*reuse_b=*/false);
    float t = d[0] + d[1] + d[2] + d[3] + d[4] + d[5] + d[6] + d[7];
    t += __shfl_xor(t, 16, 32);
    return t;
#else
    for (int off = 16; off > 0; off >>= 1) v += __shfl_xor(v, off, 32);
    return v;
#endif
}

__device__ __forceinline__ float wave_min32(float v) {
    for (int off = 16; off > 0; off >>= 1) v = fminf(v, __shfl_xor(v, off, 32));
    return v;
}
__device__ __forceinline__ float wave_max32(float v) {
    for (int off = 16; off > 0; off >>= 1) v = fmaxf(v, __shfl_xor(v, off, 32));
    return v;
}

// ---------------------------------------------------------------------------
// Pass 1: streaming copy + per-block partial reductions.
// Loads x with regular-temporal hint (keep resident in 192MB L2 across graph
// replays); writes the copy with non-temporal stores so the 128MB output
// stream does not evict x from L2.
// partials[4*b + {0,1,2,3}] = {sum, sumsq, min, max} for block b.
// ---------------------------------------------------------------------------
__global__ void __launch_bounds__(256)
stats_pass1(const float* __restrict__ x, float* __restrict__ out,
            float* __restrict__ partials, long long n) {
    const long long n4     = n >> 2;                       // # float4 elements
    const long long stride = (long long)gridDim.x * blockDim.x;
    long long i = (long long)blockIdx.x * blockDim.x + threadIdx.x;

    const v4f* __restrict__ x4   = (const v4f*)x;
    v4f* __restrict__       out4 = (v4f*)out;

    float sum = 0.0f, sq = 0.0f;
    float mn = INFINITY, mx = -INFINITY;

    // Unrolled x4: keep 4 b128 loads in flight per thread.
    for (; i + 3 * stride < n4; i += 4 * stride) {
        v4f a0 = x4[i];
        v4f a1 = x4[i + stride];
        v4f a2 = x4[i + 2 * stride];
        v4f a3 = x4[i + 3 * stride];
        __builtin_nontemporal_store(a0, &out4[i]);
        __builtin_nontemporal_store(a1, &out4[i + stride]);
        __builtin_nontemporal_store(a2, &out4[i + 2 * stride]);
        __builtin_nontemporal_store(a3, &out4[i + 3 * stride]);
#pragma unroll
        for (int k = 0; k < 4; ++k) {
            v4f a = (k == 0) ? a0 : (k == 1) ? a1 : (k == 2) ? a2 : a3;
#pragma unroll
            for (int c = 0; c < 4; ++c) {
                float v = a[c];
                sum += v;
                sq  = fmaf(v, v, sq);
                mn  = fminf(mn, v);
                mx  = fmaxf(mx, v);
            }
        }
    }
    for (; i < n4; i += stride) {
        v4f a = x4[i];
        __builtin_nontemporal_store(a, &out4[i]);
#pragma unroll
        for (int c = 0; c < 4; ++c) {
            float v = a[c];
            sum += v;
            sq  = fmaf(v, v, sq);
            mn  = fminf(mn, v);
            mx  = fmaxf(mx, v);
        }
    }
    // Scalar tail (n not divisible by 4).
    {
        long long base = n4 << 2;
        long long rem  = n - base;
        long long gid  = (long long)blockIdx.x * blockDim.x + threadIdx.x;
        if (gid < rem) {
            float v = x[base + gid];
            out[base + gid] = v;
            sum += v;
            sq  = fmaf(v, v, sq);
            mn  = fminf(mn, v);
            mx  = fmaxf(mx, v);
        }
    }

    // Wave-level reduction (WMMA for the sums), then cross-wave via LDS.
    float wsum = wave_sum32(sum);
    float wsq  = wave_sum32(sq);
    float wmn  = wave_min32(mn);
    float wmx  = wave_max32(mx);

    __shared__ float s_sum[32], s_sq[32], s_mn[32], s_mx[32];
    const int wave  = threadIdx.x >> 5;
    const int lane  = threadIdx.x & 31;
    const int nwave = blockDim.x >> 5;
    if (lane == 0) {
        s_sum[wave] = wsum; s_sq[wave] = wsq;
        s_mn[wave]  = wmn;  s_mx[wave] = wmx;
    }
    __syncthreads();
    if (threadIdx.x == 0) {
        float bs = s_sum[0], bq = s_sq[0], bmn = s_mn[0], bmx = s_mx[0];
        for (int w = 1; w < nwave; ++w) {
            bs += s_sum[w]; bq += s_sq[w];
            bmn = fminf(bmn, s_mn[w]); bmx = fmaxf(bmx, s_mx[w]);
        }
        partials[4 * blockIdx.x + 0] = bs;
        partials[4 * blockIdx.x + 1] = bq;
        partials[4 * blockIdx.x + 2] = bmn;
        partials[4 * blockIdx.x + 3] = bmx;
    }
}

// ---------------------------------------------------------------------------
// Pass 2: deterministic finalize over nb block-partials (single block).
// Double accumulation for sum/sumsq; stats order: [mean, sqmean, var, min, max].
// ---------------------------------------------------------------------------
__global__ void __launch_bounds__(256)
stats_pass2(const float* __restrict__ partials, int nb,
            float* __restrict__ out, long long n) {
    double s = 0.0, q = 0.0;
    float mn = INFINITY, mx = -INFINITY;
    for (int i = threadIdx.x; i < nb; i += blockDim.x) {
        s  += (double)partials[4 * i + 0];
        q  += (double)partials[4 * i + 1];
        mn  = fminf(mn, partials[4 * i + 2]);
        mx  = fmaxf(mx, partials[4 * i + 3]);
    }
    __shared__ double ls[256], lq[256];
    __shared__ float  lmn[256], lmx[256];
    ls[threadIdx.x] = s;  lq[threadIdx.x] = q;
    lmn[threadIdx.x] = mn; lmx[threadIdx.x] = mx;
    __syncthreads();
    if (threadIdx.x == 0) {
        double S = ls[0], Q = lq[0];
        float  MN = lmn[0], MX = lmx[0];
        for (int i = 1; i < (int)blockDim.x; ++i) {
            S += ls[i]; Q += lq[i];
            MN = fminf(MN, lmn[i]); MX = fmaxf(MX, lmx[i]);
        }
        const double N    = (double)n;
        const double mean = S / N;
        const double sqm  = Q / N;
        const double var  = (Q - S * S / N) / (N - 1.0);
        out[n + 0] = (float)mean;
        out[n + 1] = (float)sqm;
        out[n + 2] = (float)var;
        out[n + 3] = MN;
        out[n + 4] = MX;
    }
}

extern "C" void kernel_launch(void* const* d_in, const int* in_sizes, int n_in,
                              void* d_out, int out_size, void* d_ws, size_t ws_size,
                              hipStream_t stream) {
    const float* x   = (const float*)d_in[0];
    float*       out = (float*)d_out;
    float*       partials = (float*)d_ws;

    const long long n  = (long long)in_sizes[0];
    const long long n4 = n >> 2;

    const int threads = 256;
    int blocks = 2048;
    long long maxb = (n4 + threads - 1) / threads;
    if (maxb < 1) maxb = 1;
    if ((long long)blocks > maxb) blocks = (int)maxb;
    // keep partials within workspace (16 bytes per block)
    if (ws_size >= 16 && (size_t)blocks * 16 > ws_size)
        blocks = (int)(ws_size / 16);
    if (blocks < 1) blocks = 1;

    stats_pass1<<<blocks, threads, 0, stream>>>(x, out, partials, n);
    stats_pass2<<<1, threads, 0, stream>>>(partials, blocks, out, n);
}